// TralaleroTralala_68186900791671
// MI455X (gfx1250) — compile-verified
//
#include <hip/hip_runtime.h>
#include <math.h>

// ---------------------------------------------------------------------------
// Cl(3,0) geometric-algebra network, fp32, using V_WMMA_F32_16X16X4_F32.
//   B=4096, IN=512, H=32, OUT=9, D=8 blades, G=4 grades.
// Blade index ordering (grade-major): bitmaps {0,1,2,4,3,5,6,7}.
//
// Pipeline:
//   k0a/k0b/k0c : one-time weight transposes -> lane-contiguous B-operands
//   k1          : xr1 = normalization(mv_linear(x, w_right1))        [WMMA]
//   k2          : h   = silu((left + geometric-product) / sqrt2)     [WMMA]
//   k3          : layers 2+3 tail, one wave per batch element        [VALU]
// ---------------------------------------------------------------------------

typedef float vf2 __attribute__((ext_vector_type(2)));
typedef float vf8 __attribute__((ext_vector_type(8)));

#define WMMA_F32(a, b, c) \
  __builtin_amdgcn_wmma_f32_16x16x4_f32(false, (a), false, (b), (short)0, (c), false, false)

static constexpr int   BM_c[8] = {0, 1, 2, 4, 3, 5, 6, 7};  // index -> bitmap
static constexpr int   IDX_c[8] = {0, 1, 2, 4, 3, 5, 6, 7}; // bitmap -> index (self-inverse)
static constexpr int   GR_c[8] = {0, 1, 1, 1, 2, 2, 2, 3};  // index -> grade
static constexpr float INV_SQRT2_F = 0.70710678118654752440f;
static constexpr float EPS_F = 1e-6f;

__host__ __device__ constexpr float blade_sign(int a, int b) {
  int s = 1;
  int aa = a >> 1;
  while (aa) {
    int v = aa & b;
    int pc = (v & 1) + ((v >> 1) & 1) + ((v >> 2) & 1);
    if (pc & 1) s = -s;
    aa >>= 1;
  }
  return (float)s;  // METRIC = (1,1,1): no metric sign
}

__device__ __forceinline__ float sigmoidf_(float v) { return 1.0f / (1.0f + __expf(-v)); }

// normalization(): y / (sigmoid(a)*(sqrt(|q_g|)-1)+1 + eps), per-grade
__device__ __forceinline__ void norm_apply(const float* y, const float* na, float* o) {
  float q0 = y[0] * y[0];
  float q1 = y[1] * y[1] + y[2] * y[2] + y[3] * y[3];
  float q2 = y[4] * y[4] + y[5] * y[5] + y[6] * y[6];
  float q3 = y[7] * y[7];
  float4 a = *(const float4*)na;
  float nrm[4];
  nrm[0] = sigmoidf_(a.x) * (sqrtf(fabsf(q0)) - 1.0f) + 1.0f;
  nrm[1] = sigmoidf_(a.y) * (sqrtf(fabsf(q1)) - 1.0f) + 1.0f;
  nrm[2] = sigmoidf_(a.z) * (sqrtf(fabsf(q2)) - 1.0f) + 1.0f;
  nrm[3] = sigmoidf_(a.w) * (sqrtf(fabsf(q3)) - 1.0f) + 1.0f;
#pragma unroll
  for (int ii = 0; ii < 8; ++ii) o[ii] = y[ii] / (nrm[GR_c[ii]] + EPS_F);
}

// mv_silu(): gate = sigmoid(a*inv + b), inv = {x0, q1, q2, q3}
__device__ __forceinline__ void silu_apply(float* h, const float* A, const float* Bb) {
  float q1 = h[1] * h[1] + h[2] * h[2] + h[3] * h[3];
  float q2 = h[4] * h[4] + h[5] * h[5] + h[6] * h[6];
  float q3 = h[7] * h[7];
  float4 a = *(const float4*)A;
  float4 b = *(const float4*)Bb;
  float g[4];
  g[0] = sigmoidf_(a.x * h[0] + b.x);
  g[1] = sigmoidf_(a.y * q1 + b.y);
  g[2] = sigmoidf_(a.z * q2 + b.z);
  g[3] = sigmoidf_(a.w * q3 + b.w);
#pragma unroll
  for (int ii = 0; ii < 8; ++ii) h[ii] *= g[GR_c[ii]];
}

// ---------------------------------------------------------------------------
// Async global->LDS staging (CDNA5 GLOBAL_LOAD_ASYNC_TO_LDS_B128, ASYNCcnt),
// with a compile-safe synchronous fallback.
// The builtin's param 0 is `v4i addrspace(1)*` (revealed by clang diagnostic),
// param 1 the LDS destination, then imm offset + imm cpol.
// ---------------------------------------------------------------------------
#if defined(__has_builtin)
#if __has_builtin(__builtin_amdgcn_global_load_async_to_lds_b128) && \
    __has_builtin(__builtin_amdgcn_s_wait_asynccnt)
#define HAVE_ASYNC_LDS 1
#endif
#endif

typedef int async_v4i __attribute__((vector_size(16)));
typedef __attribute__((address_space(1))) async_v4i* async_gptr;
typedef __attribute__((address_space(3))) async_v4i* async_lptr;

__device__ __forceinline__ void stage16(const float* __restrict__ gp, float* lp) {
#if defined(HAVE_ASYNC_LDS)
  __builtin_amdgcn_global_load_async_to_lds_b128((async_gptr)gp, (async_lptr)lp, 0, 0);
#else
  *(float4*)lp = *(const float4*)gp;
#endif
}
__device__ __forceinline__ void stage_wait() {
#if defined(HAVE_ASYNC_LDS)
  __builtin_amdgcn_s_wait_asynccnt(0);
#endif
}

// ---------------------------------------------------------------------------
// Kernel 0: weight transposes so GEMM B-operand rows are lane-contiguous.
//   wr1T[g][m][n]  = w_right1[n][m][g]   (4 x 512 x 512)
//   gpT [t][n][m]  = gp_w1   [m][n][t]   (64 x 512 x 32)
//   wlT [g][n][m]  = w_left1 [m][n][g]   (4 x 512 x 32)
// Writes coalesced; scattered reads are a one-time ~17 MB cost.
// ---------------------------------------------------------------------------
__global__ __launch_bounds__(256) void k0_wr1T(const float* __restrict__ wr1,
                                               float* __restrict__ wr1T) {
  int e = blockIdx.x * 256 + threadIdx.x;  // 4*512*512 elements
  int n = e & 511, m = (e >> 9) & 511, g = e >> 18;
  wr1T[e] = wr1[((size_t)n * 512 + m) * 4 + g];
}
__global__ __launch_bounds__(256) void k0_gpT(const float* __restrict__ gpw1,
                                              float* __restrict__ gpT) {
  int e = blockIdx.x * 256 + threadIdx.x;  // 64*512*32 elements
  int m = e & 31, n = (e >> 5) & 511, t = e >> 14;
  gpT[e] = gpw1[((size_t)m * 512 + n) * 64 + t];
}
__global__ __launch_bounds__(256) void k0_wlT(const float* __restrict__ wl1,
                                              float* __restrict__ wlT) {
  int e = blockIdx.x * 256 + threadIdx.x;  // 4*512*32 elements
  int m = e & 31, n = (e >> 5) & 511, g = e >> 14;
  wlT[e] = wl1[((size_t)m * 512 + n) * 4 + g];
}

// ---------------------------------------------------------------------------
// Kernel 1: xr1 = normalization(mv_linear(x, w_right1), norm_a1)
//   Per WG: 16b x 16n output tile, 8 waves = one per blade component.
//   x tile async-staged to LDS; 128 f32 WMMAs per wave.
// ---------------------------------------------------------------------------
__global__ __launch_bounds__(256) void k1_right_norm(const float* __restrict__ x,
                                                     const float* __restrict__ wr1T,
                                                     const float* __restrict__ na1,
                                                     float* __restrict__ xr) {
  __shared__ float lds[16 * 16 * 8];  // 8 KB: x stage, then y exchange
  const int tid = threadIdx.x;
  const int lane = tid & 31;
  const int i = tid >> 5;        // blade component handled by this wave
  const int g = GR_c[i];         // its grade -> weight plane
  const int b0 = blockIdx.x * 16;
  const int n0 = blockIdx.y * 16;
  const int blS = tid >> 4, mlS = tid & 15;  // staging coords
  const int M = lane & 15, Kb = (lane >> 4) * 2;

  vf8 c = {0.f, 0.f, 0.f, 0.f, 0.f, 0.f, 0.f, 0.f};

  for (int mc = 0; mc < 512; mc += 16) {
    // stage x[b0+bl][mc+ml][0..7]  (32B per thread, fully coalesced)
    const float* src = x + ((size_t)(b0 + blS) * 512 + (mc + mlS)) * 8;
    float* dst = &lds[(blS * 16 + mlS) * 8];
    stage16(src, dst);
    stage16(src + 4, dst + 4);
    if (mc + 16 < 512)
      __builtin_prefetch(x + ((size_t)(b0 + blS) * 512 + (mc + 16 + mlS)) * 8, 0, 1);
    stage_wait();
    __syncthreads();

#pragma unroll
    for (int kk = 0; kk < 4; ++kk) {
      int k = kk * 4 + Kb;
      // A (16x4): lane L<16 -> K={k,k+1} rows for M=L; L>=16 -> K+2
      vf2 a;
      a.x = lds[(M * 16 + k) * 8 + i];
      a.y = lds[(M * 16 + k + 1) * 8 + i];
      // B (4x16): wr1T[g][m][n] -> lane-contiguous over n
      const float* wp = wr1T + ((size_t)g * 512 + (mc + k)) * 512 + n0 + M;
      vf2 bv;
      bv.x = wp[0];
      bv.y = wp[512];
      c = WMMA_F32(a, bv, c);
    }
    __syncthreads();
  }

  // exchange all 8 components through LDS for fused normalization
#pragma unroll
  for (int r = 0; r < 8; ++r) {
    int bb = r + (lane >> 4) * 8;  // C/D layout: VGPR r holds rows r and r+8
    int nn = lane & 15;
    lds[(bb * 16 + nn) * 8 + i] = c[r];
  }
  __syncthreads();

  {
    const int bb = tid >> 4, nn = tid & 15;  // one (b,n) pair per thread
    float yv[8];
#pragma unroll
    for (int ii = 0; ii < 8; ++ii) yv[ii] = lds[(bb * 16 + nn) * 8 + ii];
    float o[8];
    norm_apply(yv, na1 + (size_t)(n0 + nn) * 4, o);
    float4* dst = (float4*)(xr + ((size_t)(b0 + bb) * 512 + (n0 + nn)) * 8);
    float4 v0, v1;
    v0.x = o[0]; v0.y = o[1]; v0.z = o[2]; v0.w = o[3];
    v1.x = o[4]; v1.y = o[5]; v1.z = o[6]; v1.w = o[7];
    dst[0] = v0;
    dst[1] = v1;
  }
}

// ---------------------------------------------------------------------------
// Kernel 2: h = mv_silu((mv_linear(x,w_left1,b) + GP(x, xr1; gp_w1)) * 1/sqrt2)
//   Per WG: 16b x (32m x 8j) tile; 8 waves = one per output blade j.
//   Each wave: 9 GEMM chains (8 Cayley pairs + left linear) over n=512,
//   A-operand s * x_i * xr_k built from LDS; two 16x16 WMMA accumulators.
//   B-operands from transposed weights: lane-contiguous 128B rows.
// ---------------------------------------------------------------------------
__global__ __launch_bounds__(256) void k2_gp_left_silu(const float* __restrict__ x,
                                                       const float* __restrict__ xr,
                                                       const float* __restrict__ gpT,
                                                       const float* __restrict__ wlT,
                                                       const float* __restrict__ bl1,
                                                       const float* __restrict__ sa,
                                                       const float* __restrict__ sb,
                                                       float* __restrict__ hb) {
  __shared__ float lds[4096];  // 16 KB: [0..2047]=x, [2048..4095]=xr; reused for h
  const int tid = threadIdx.x;
  const int lane = tid & 31;
  const int j = tid >> 5;  // output blade component
  const int gj = GR_c[j];

  // pair table for this j:  i^k = j  =>  k = IDX[BM[i]^BM[j]]
  int ib_t[8], tt_t[8];
  float ss_t[8];
#pragma unroll
  for (int p = 0; p < 8; ++p) {
    int bma = BM_c[p];
    int bmb = bma ^ BM_c[j];
    int ib = IDX_c[bmb];
    ib_t[p] = ib;
    ss_t[p] = blade_sign(bma, bmb);
    tt_t[p] = (GR_c[p] * 4 + gj) * 4 + GR_c[ib];
  }

  const int b0 = blockIdx.x * 16;
  const int blS = tid >> 4, nlS = tid & 15;
  const int M = lane & 15, Kb = (lane >> 4) * 2, N = lane & 15;

  vf8 c0 = {0.f, 0.f, 0.f, 0.f, 0.f, 0.f, 0.f, 0.f};
  vf8 c1 = {0.f, 0.f, 0.f, 0.f, 0.f, 0.f, 0.f, 0.f};

  for (int nc = 0; nc < 512; nc += 16) {
    size_t gidx = ((size_t)(b0 + blS) * 512 + nc + nlS) * 8;
    float* dx = &lds[(blS * 16 + nlS) * 8];
    float* dr = &lds[2048 + (blS * 16 + nlS) * 8];
    stage16(x + gidx, dx);
    stage16(x + gidx + 4, dx + 4);
    stage16(xr + gidx, dr);
    stage16(xr + gidx + 4, dr + 4);
    stage_wait();
    __syncthreads();

#pragma unroll
    for (int p = 0; p < 9; ++p) {
#pragma unroll
      for (int kk = 0; kk < 4; ++kk) {
        int k = kk * 4 + Kb;
        vf2 a, bv0, bv1;
        if (p < 8) {
          const int ia = p, ib = ib_t[p], t = tt_t[p];
          const float s = ss_t[p];
          a.x = s * lds[(M * 16 + k) * 8 + ia] * lds[2048 + (M * 16 + k) * 8 + ib];
          a.y = s * lds[(M * 16 + k + 1) * 8 + ia] * lds[2048 + (M * 16 + k + 1) * 8 + ib];
          // gpT[t][n][m]: rows of 32 m's, lane-contiguous
          const float* w = gpT + ((size_t)t * 512 + (nc + k)) * 32;
          bv0.x = w[N];       bv0.y = w[32 + N];
          bv1.x = w[N + 16];  bv1.y = w[48 + N];
        } else {  // left linear: A = x component j, W = wlT[gj][n][m]
          a.x = lds[(M * 16 + k) * 8 + j];
          a.y = lds[(M * 16 + k + 1) * 8 + j];
          const float* w = wlT + ((size_t)gj * 512 + (nc + k)) * 32;
          bv0.x = w[N];       bv0.y = w[32 + N];
          bv1.x = w[N + 16];  bv1.y = w[48 + N];
        }
        c0 = WMMA_F32(a, bv0, c0);
        c1 = WMMA_F32(a, bv1, c1);
      }
    }
    __syncthreads();
  }

  // bias (component 0 only) + 1/sqrt2
  const float add0 = (j == 0) ? bl1[N] : 0.f;
  const float add1 = (j == 0) ? bl1[N + 16] : 0.f;
#pragma unroll
  for (int r = 0; r < 8; ++r) {
    c0[r] = (c0[r] + add0) * INV_SQRT2_F;
    c1[r] = (c1[r] + add1) * INV_SQRT2_F;
  }

  // exchange to LDS as h[b][m][j] for fused mv_silu
#pragma unroll
  for (int r = 0; r < 8; ++r) {
    int bb = r + (lane >> 4) * 8;
    lds[(bb * 32 + N) * 8 + j] = c0[r];
    lds[(bb * 32 + (N + 16)) * 8 + j] = c1[r];
  }
  __syncthreads();

#pragma unroll
  for (int q = 0; q < 2; ++q) {
    int e = tid + q * 256;  // 512 (b,m) pairs
    int bb = e >> 5, m = e & 31;
    float hv[8];
#pragma unroll
    for (int ii = 0; ii < 8; ++ii) hv[ii] = lds[(bb * 32 + m) * 8 + ii];
    silu_apply(hv, sa + m * 4, sb + m * 4);
    float4* dst = (float4*)(hb + ((size_t)(b0 + bb) * 32 + m) * 8);
    float4 v0, v1;
    v0.x = hv[0]; v0.y = hv[1]; v0.z = hv[2]; v0.w = hv[3];
    v1.x = hv[4]; v1.y = hv[5]; v1.z = hv[6]; v1.w = hv[7];
    dst[0] = v0;
    dst[1] = v1;
  }
}

// ---------------------------------------------------------------------------
// Kernel 3: tail (sgp layer 2 + silu + fcgp layer 3) — one wave per batch b,
//   lane = channel n (H=32 == warpSize). ~0.3 GFLOP total, pure VALU.
// ---------------------------------------------------------------------------
#define WSEL(w4, ii) ((ii) == 0 ? (w4).x : ((ii) < 4 ? (w4).y : ((ii) < 7 ? (w4).z : (w4).w)))

__global__ __launch_bounds__(256) void k3_tail(const float* __restrict__ hb,
                                               const float* __restrict__ wr2,
                                               const float* __restrict__ na2,
                                               const float* __restrict__ wl2,
                                               const float* __restrict__ bl2,
                                               const float* __restrict__ gpw2,
                                               const float* __restrict__ sa,
                                               const float* __restrict__ sb,
                                               const float* __restrict__ wr3,
                                               const float* __restrict__ na3,
                                               const float* __restrict__ wl3,
                                               const float* __restrict__ bl3,
                                               const float* __restrict__ gpw3,
                                               float* __restrict__ out) {
  const int tid = threadIdx.x;
  const int n = tid & 31;                     // channel = lane (wave32)
  const int b = blockIdx.x * 8 + (tid >> 5);  // 8 waves per block

  float h_[8];
  {
    const float4* hp = (const float4*)(hb + ((size_t)b * 32 + n) * 8);
    float4 h0 = hp[0], h1 = hp[1];
    h_[0] = h0.x; h_[1] = h0.y; h_[2] = h0.z; h_[3] = h0.w;
    h_[4] = h1.x; h_[5] = h1.y; h_[6] = h1.z; h_[7] = h1.w;
  }

  // ---- layer 2: right linear (32x32 per grade) via lane broadcast ----
  float y_[8] = {0, 0, 0, 0, 0, 0, 0, 0};
  for (int m = 0; m < 32; ++m) {
    float4 w4 = *(const float4*)(wr2 + ((size_t)n * 32 + m) * 4);
#pragma unroll
    for (int ii = 0; ii < 8; ++ii) y_[ii] += __shfl(h_[ii], m, 32) * WSEL(w4, ii);
  }
  float xr2_[8];
  norm_apply(y_, na2 + (size_t)n * 4, xr2_);

  // ---- sgp: per-channel geometric product + left linear ----
  float h2_[8];
#pragma unroll
  for (int jj = 0; jj < 8; ++jj) {
    float acc = 0.f;
#pragma unroll
    for (int ia = 0; ia < 8; ++ia) {
      const int bma = BM_c[ia], bmb = bma ^ BM_c[jj];
      const int ib = IDX_c[bmb];
      const float s = blade_sign(bma, bmb);
      const int t = (GR_c[ia] * 4 + GR_c[jj]) * 4 + GR_c[ib];
      acc += s * h_[ia] * xr2_[ib] * gpw2[(size_t)n * 64 + t];
    }
    h2_[jj] = acc;
  }
  {
    float l_[8] = {0, 0, 0, 0, 0, 0, 0, 0};
    for (int m = 0; m < 32; ++m) {
      float4 w4 = *(const float4*)(wl2 + ((size_t)n * 32 + m) * 4);
#pragma unroll
      for (int ii = 0; ii < 8; ++ii) l_[ii] += __shfl(h_[ii], m, 32) * WSEL(w4, ii);
    }
    l_[0] += bl2[n];
#pragma unroll
    for (int ii = 0; ii < 8; ++ii) h2_[ii] = (l_[ii] + h2_[ii]) * INV_SQRT2_F;
  }
  silu_apply(h2_, sa + (size_t)n * 4, sb + (size_t)n * 4);

  // ---- layer 3: right linear + normalization ----
  float y3_[8] = {0, 0, 0, 0, 0, 0, 0, 0};
  for (int m = 0; m < 32; ++m) {
    float4 w4 = *(const float4*)(wr3 + ((size_t)n * 32 + m) * 4);
#pragma unroll
    for (int ii = 0; ii < 8; ++ii) y3_[ii] += __shfl(h2_[ii], m, 32) * WSEL(w4, ii);
  }
  float xr3_[8];
  norm_apply(y3_, na3 + (size_t)n * 4, xr3_);

  // ---- output fcgp: reduce over channels with butterfly shuffles ----
  for (int o = 0; o < 9; ++o) {
#pragma unroll
    for (int jj = 0; jj < 8; ++jj) {
      float acc = h2_[jj] * wl3[((size_t)o * 32 + n) * 4 + GR_c[jj]];
#pragma unroll
      for (int ia = 0; ia < 8; ++ia) {
        const int bma = BM_c[ia], bmb = bma ^ BM_c[jj];
        const int ib = IDX_c[bmb];
        const float s = blade_sign(bma, bmb);
        const int t = (GR_c[ia] * 4 + GR_c[jj]) * 4 + GR_c[ib];
        acc += s * h2_[ia] * xr3_[ib] * gpw3[((size_t)o * 32 + n) * 64 + t];
      }
#pragma unroll
      for (int off = 16; off >= 1; off >>= 1) acc += __shfl_xor(acc, off, 32);
      if (n == 0) out[((size_t)b * 9 + o) * 8 + jj] = (acc + ((jj == 0) ? bl3[o] : 0.f)) * INV_SQRT2_F;
    }
  }
}

// ---------------------------------------------------------------------------
extern "C" void kernel_launch(void* const* d_in, const int* in_sizes, int n_in,
                              void* d_out, int out_size, void* d_ws, size_t ws_size,
                              hipStream_t stream) {
  (void)in_sizes; (void)n_in; (void)out_size; (void)ws_size;
  const float* x    = (const float*)d_in[0];
  const float* wr1  = (const float*)d_in[1];
  const float* na1  = (const float*)d_in[2];
  const float* wl1  = (const float*)d_in[3];
  const float* bl1  = (const float*)d_in[4];
  const float* gpw1 = (const float*)d_in[5];
  const float* sa   = (const float*)d_in[6];
  const float* sb   = (const float*)d_in[7];
  const float* wr2  = (const float*)d_in[8];
  const float* na2  = (const float*)d_in[9];
  const float* wl2  = (const float*)d_in[10];
  const float* bl2  = (const float*)d_in[11];
  const float* gpw2 = (const float*)d_in[12];
  const float* wr3  = (const float*)d_in[13];
  const float* na3  = (const float*)d_in[14];
  const float* wl3  = (const float*)d_in[15];
  const float* bl3  = (const float*)d_in[16];
  const float* gpw3 = (const float*)d_in[17];
  float* out = (float*)d_out;

  // workspace: xr1 (64 MB) | h (4 MB) | wr1T (4 MB) | gpT (4 MB) | wlT (256 KB)
  float* xr   = (float*)d_ws;
  float* hb   = xr + (size_t)4096 * 512 * 8;
  float* wr1T = hb + (size_t)4096 * 32 * 8;
  float* gpT  = wr1T + (size_t)4 * 512 * 512;
  float* wlT  = gpT + (size_t)64 * 512 * 32;

  k0_wr1T<<<(4 * 512 * 512) / 256, 256, 0, stream>>>(wr1, wr1T);
  k0_gpT<<<(64 * 512 * 32) / 256, 256, 0, stream>>>(gpw1, gpT);
  k0_wlT<<<(4 * 512 * 32) / 256, 256, 0, stream>>>(wl1, wlT);

  dim3 g1(4096 / 16, 512 / 16, 1);
  k1_right_norm<<<g1, 256, 0, stream>>>(x, wr1T, na1, xr);

  dim3 g2(4096 / 16, 1, 1);
  k2_gp_left_silu<<<g2, 256, 0, stream>>>(x, xr, gpT, wlT, bl1, sa, sb, hb);

  dim3 g3(4096 / 8, 1, 1);
  k3_tail<<<g3, 256, 0, stream>>>(hb, wr2, na2, wl2, bl2, gpw2, sa, sb,
                                  wr3, na3, wl3, bl3, gpw3, out);
}